// LDM_34806414967248
// MI455X (gfx1250) — compile-verified
//
#include <hip/hip_runtime.h>
#include <math.h>

// ---------------------------------------------------------------------------
// LDM Kalman forward pass, MI455X (gfx1250), wave32, one wave per sequence.
// Information-form update (algebraically identical to reference for binary
// mask):   M = Lam^-1 + C^T R^-1 C  =  Lam^-1 (I + Lam G)
//   =>     Lam_t = M^-1 = (I + Lam G)^-1 Lam   (single 16x16 solve)
//          K     = M^-1 C^T R^-1               (WMMA)
// P = Lam@G + I is built on v_wmma_f32_16x16x4_f32; the solve is a fully
// unrolled register-resident Gauss-Jordan with v_readlane pivot broadcasts.
// ---------------------------------------------------------------------------

typedef __attribute__((ext_vector_type(2))) float v2f;
typedef __attribute__((ext_vector_type(8))) float v8f;

#define T_STEPS 1024
#define NSEQ    256
#define DX      16
#define DU      16
#define DA      32
#define WPB     8      // waves (sequences) per block
#define S16     17     // LDS row stride for 16-wide arrays
#define S32     34     // LDS row stride for 32-wide arrays

// output offsets (floats): (mu_pred, mu_t, Lambda_pred, Lambda_t), time-major
#define OFF_MUP  ((size_t)0)
#define OFF_MUT  ((size_t)T_STEPS * NSEQ * DX)
#define OFF_LAMP ((size_t)2 * T_STEPS * NSEQ * DX)
#define OFF_LAMT (OFF_LAMP + (size_t)T_STEPS * NSEQ * DX * DX)

__device__ __forceinline__ void lds_sync() {
  asm volatile("s_wait_dscnt 0" ::: "memory");
}

__device__ __forceinline__ float rdlane_f(float v, int l) {
  return __int_as_float(__builtin_amdgcn_readlane(__float_as_int(v), l));
}

// v_rcp_f32 + one Newton iteration (plenty for the well-conditioned pivots)
__device__ __forceinline__ float fast_rcp(float x) {
  float r = __builtin_amdgcn_rcpf(x);
  return r * fmaf(-x, r, 2.0f);
}

// --- WMMA f32 16x16x4 fragment helpers (layouts per CDNA5 ISA) ---
__device__ __forceinline__ void load_a(v2f af[4], const float* p, int stride,
                                       int ln, int hl) {
#pragma unroll
  for (int kk = 0; kk < 4; ++kk)
#pragma unroll
    for (int j = 0; j < 2; ++j)
      af[kk][j] = p[ln * stride + kk * 4 + j + 2 * hl];
}
__device__ __forceinline__ void load_b(v2f bf[4], const float* p, int stride,
                                       int ln, int hl) {
#pragma unroll
  for (int kk = 0; kk < 4; ++kk)
#pragma unroll
    for (int j = 0; j < 2; ++j)
      bf[kk][j] = p[(kk * 4 + j + 2 * hl) * stride + ln];
}
// Transposed B: element (K,N) = p[N*stride + K] (same addressing as load_a)
__device__ __forceinline__ void load_bt(v2f bf[4], const float* p, int stride,
                                        int ln, int hl) {
  load_a(bf, p, stride, ln, hl);
}
__device__ __forceinline__ void store_cd(float* p, int stride, v8f d, int ln, int hl) {
#pragma unroll
  for (int j = 0; j < 8; ++j) p[(j + 8 * hl) * stride + ln] = d[j];
}
__device__ __forceinline__ void store_cd_g(float* __restrict__ p, v8f d, int ln, int hl) {
#pragma unroll
  for (int j = 0; j < 8; ++j) p[(j + 8 * hl) * 16 + ln] = d[j];
}
__device__ __forceinline__ v8f wmma16(const v2f a[4], const v2f b[4], v8f c) {
#pragma unroll
  for (int kk = 0; kk < 4; ++kk)
    c = __builtin_amdgcn_wmma_f32_16x16x4_f32(false, a[kk], false, b[kk],
                                              (short)0, c, false, false);
  return c;
}

// Gauss-Jordan solve on [P | RHS] (16x32), one row per lane (lanes 0..15
// meaningful), no row scaling.  Live columns at pivot p: P-cols >p plus the
// dense 16-col RHS.  Returns per-lane pivot d_i; solution row i of P^-1*RHS
// is rowv[16..31] * (1/d_i).
__device__ __forceinline__ float gjsolve16(float rowv[32], int lane) {
  float dval = 1.0f;
#pragma unroll
  for (int p = 0; p < 16; ++p) {
    float piv = rdlane_f(rowv[p], p);
    float ipv = fast_rcp(piv);
    float f = (lane == p) ? 0.0f : rowv[p] * ipv;
    dval = (lane == p) ? piv : dval;
#pragma unroll
    for (int c = p + 1; c < 16; ++c)
      rowv[c] = fmaf(-f, rdlane_f(rowv[c], p), rowv[c]);
#pragma unroll
    for (int c = 16; c < 32; ++c)
      rowv[c] = fmaf(-f, rdlane_f(rowv[c], p), rowv[c]);
  }
  return dval;
}

__global__ void __launch_bounds__(WPB * 32)
ldm_forward_kernel(const float* __restrict__ a_in, const float* __restrict__ u_in,
                   const float* __restrict__ mask_in, const float* __restrict__ A_in,
                   const float* __restrict__ B_in, const float* __restrict__ C_in,
                   const float* __restrict__ mu0_in, const float* __restrict__ Lam0_in,
                   const float* __restrict__ wlog_in, const float* __restrict__ rlog_in,
                   float* __restrict__ out) {
  // block-wide constants
  __shared__ float sA[DX * S16];     // A
  __shared__ float sB[DX * S16];     // B
  __shared__ float sC[DA * S16];     // C (rows a, cols x)
  __shared__ float sCtR[DX * S32];   // C^T R^-1   [x][a]
  __shared__ float sCRC[DX * S16];   // G = C^T R^-1 C [x][y]
  // per-wave scratch
  __shared__ float sLam[WPB][DX * S16];   // Lambda_pred (state)
  __shared__ float sLamT[WPB][DX * S16];  // Lambda_t
  __shared__ float sT1[WPB][DX * S16];    // A @ Lambda_t
  __shared__ float sMinv[WPB][DX * S16];  // P, then M^-1
  __shared__ float sK[WPB][DX * S32];     // masked Kalman gain

  const int tid  = threadIdx.x;
  const int lane = tid & 31;
  const int wv   = tid >> 5;
  const int hl   = lane >> 4;
  const int ln   = lane & 15;
  const int seq  = blockIdx.x * WPB + wv;

  // constants -> LDS (one-time, cooperative)
  for (int i = tid; i < DX * DX; i += blockDim.x) {
    sA[(i >> 4) * S16 + (i & 15)] = A_in[i];
    sB[(i >> 4) * S16 + (i & 15)] = B_in[i];
  }
  for (int i = tid; i < DA * DX; i += blockDim.x)
    sC[(i >> 4) * S16 + (i & 15)] = C_in[i];
  for (int i = tid; i < DX * DA; i += blockDim.x) {
    int x = i >> 5, aa = i & 31;
    sCtR[x * S32 + aa] = C_in[aa * DX + x] * expf(-rlog_in[aa]);
  }
  for (int i = tid; i < DX * DX; i += blockDim.x) {
    int x = i >> 4, y = i & 15;
    float acc = 0.f;
    for (int aa = 0; aa < DA; ++aa)
      acc = fmaf(C_in[aa * DX + x] * expf(-rlog_in[aa]), C_in[aa * DX + y], acc);
    sCRC[x * S16 + y] = acc;
  }
  __syncthreads();

  float* lam  = sLam[wv];
  float* lamt = sLamT[wv];
  float* t1m  = sT1[wv];
  float* minv = sMinv[wv];
  float* km   = sK[wv];

  // state init
  for (int e = lane; e < DX * DX; e += 32)
    lam[(e >> 4) * S16 + (e & 15)] = Lam0_in[e];
  float mu_pred = mu0_in[ln];  // lanes 16..31 hold duplicates
  lds_sync();

  // diagonal fragments: identity, W = diag(exp(wlog))
  v8f zf = {0.f, 0.f, 0.f, 0.f, 0.f, 0.f, 0.f, 0.f};
  v8f wfrag = zf, ifrag = zf;
#pragma unroll
  for (int j = 0; j < 8; ++j) {
    int M = j + 8 * hl;
    if (M == ln) {
      wfrag[j] = expf(wlog_in[M]);
      ifrag[j] = 1.0f;
    }
  }

#pragma unroll 1
  for (int t = 0; t < T_STEPS; ++t) {
    float a_val = a_in[((size_t)seq * T_STEPS + t) * DA + lane];
    float mval  = mask_in[(size_t)seq * T_STEPS + t];
    float u_val = (t == T_STEPS - 1) ? 0.f
                  : u_in[((size_t)seq * T_STEPS + t) * DU + ln];

    // ---- P = Lam_pred @ G + I   (WMMA) ----
    {
      v2f afL[4], bfG[4];
      load_a(afL, lam, S16, ln, hl);
      load_b(bfG, sCRC, S16, ln, hl);
      v8f d = wmma16(afL, bfG, ifrag);
      store_cd(minv, S16, d, ln, hl);   // stage P in the minv buffer
    }
    lds_sync();

    // residual r[lane] = a*mask - (C mu_pred)[lane]
    float ap = 0.f;
#pragma unroll
    for (int k = 0; k < DX; ++k)
      ap = fmaf(sC[lane * S16 + k], rdlane_f(mu_pred, k), ap);
    float r = a_val * mval - ap;

    // ---- solve P X = Lam_pred  =>  X = M^-1 (= unmasked Lam_t) ----
    float rowv[32];
#pragma unroll
    for (int c = 0; c < 16; ++c) rowv[c] = minv[ln * S16 + c];       // P row
#pragma unroll
    for (int c = 0; c < 16; ++c) rowv[16 + c] = lam[ln * S16 + c];   // RHS row
    float d1  = gjsolve16(rowv, lane);
    float di1 = fast_rcp(d1);

    // ---- Minv rows -> LDS (overwrite P); Lam_t = Lam + m*(Minv - Lam)
    //      (exact for binary mask); coalesced b128 global stores of Lam_t
    if (lane < 16) {
      float buf[16];
#pragma unroll
      for (int c = 0; c < 16; ++c) {
        float mv = rowv[16 + c] * di1;
        minv[ln * S16 + c] = mv;
        float lp = lam[ln * S16 + c];
        float lt = fmaf(mval, mv - lp, lp);
        lamt[ln * S16 + c] = lt;
        buf[c] = lt;
      }
      float4* gp = (float4*)(out + OFF_LAMT +
                             ((size_t)t * NSEQ + seq) * (DX * DX) + ln * DX);
#pragma unroll
      for (int q = 0; q < 4; ++q) {
        float4 v; v.x = buf[4*q]; v.y = buf[4*q+1]; v.z = buf[4*q+2]; v.w = buf[4*q+3];
        gp[q] = v;
      }
    }
    lds_sync();

    // ---- K = Minv @ (C^T R^-1), masked   [16x32]  (WMMA) ----
    {
      v2f afM[4];
      load_a(afM, minv, S16, ln, hl);
#pragma unroll
      for (int jt = 0; jt < 2; ++jt) {
        v2f bf[4];
        load_b(bf, sCtR + jt * 16, S32, ln, hl);
        v8f d = wmma16(afM, bf, zf);
        store_cd(km + jt * 16, S32, d * mval, ln, hl);
      }
    }
    lds_sync();

    // ---- mu_t = mu_pred + K r  (matches reference masked-K formula) ----
    float acc = 0.f;
#pragma unroll
    for (int aa = 0; aa < DA; ++aa)
      acc = fmaf(km[ln * S32 + aa], rdlane_f(r, aa), acc);
    float mu_t = mu_pred + acc;

    // ---- T1 = A @ Lam_t   (WMMA) ----
    {
      v2f afA[4], bfLt[4];
      load_a(afA, sA, S16, ln, hl);
      load_b(bfLt, lamt, S16, ln, hl);
      v8f d = wmma16(afA, bfLt, zf);
      store_cd(t1m, S16, d, ln, hl);
    }
    lds_sync();

    // ---- Lam_pred' = T1 @ A^T + W   (WMMA) ----
    {
      v2f afT[4], bfAT[4];
      load_a(afT, t1m, S16, ln, hl);
      load_bt(bfAT, sA, S16, ln, hl);
      v8f d = wmma16(afT, bfAT, wfrag);
      store_cd(lam, S16, d, ln, hl);   // state for next step
      store_cd_g(out + OFF_LAMP + ((size_t)t * NSEQ + seq) * (DX * DX),
                 d, ln, hl);
    }
    lds_sync();

    // ---- mu_pred' = A mu_t + B u ----
    float mp = 0.f;
#pragma unroll
    for (int k = 0; k < DX; ++k) {
      mp = fmaf(sA[ln * S16 + k], rdlane_f(mu_t, k), mp);
      mp = fmaf(sB[ln * S16 + k], rdlane_f(u_val, k), mp);
    }
    mu_pred = mp;

    if (lane < DX) {
      out[OFF_MUT + ((size_t)t * NSEQ + seq) * DX + lane] = mu_t;
      out[OFF_MUP + ((size_t)t * NSEQ + seq) * DX + lane] = mu_pred;
    }
  }
}

extern "C" void kernel_launch(void* const* d_in, const int* in_sizes, int n_in,
                              void* d_out, int out_size, void* d_ws, size_t ws_size,
                              hipStream_t stream) {
  (void)in_sizes; (void)n_in; (void)out_size; (void)d_ws; (void)ws_size;
  const float* a    = (const float*)d_in[0];
  const float* u    = (const float*)d_in[1];
  const float* mask = (const float*)d_in[2];
  const float* A    = (const float*)d_in[3];
  const float* B    = (const float*)d_in[4];
  const float* C    = (const float*)d_in[5];
  const float* mu0  = (const float*)d_in[6];
  const float* Lam0 = (const float*)d_in[7];
  const float* wlog = (const float*)d_in[8];
  const float* rlog = (const float*)d_in[9];
  float* out = (float*)d_out;

  dim3 grid(NSEQ / WPB);   // 32 blocks
  dim3 block(WPB * 32);    // 256 threads = 8 waves
  ldm_forward_kernel<<<grid, block, 0, stream>>>(a, u, mask, A, B, C, mu0, Lam0,
                                                 wlog, rlog, out);
}